// LorentzBatchNormLVar_21199958573139
// MI455X (gfx1250) — compile-verified
//
#include <hip/hip_runtime.h>
#include <stdint.h>

#define DLEN 512
#define EPSV 1e-5f
#define MIN_NORM 1e-8f
// arcsinh(sqrt(fp32_max)) = ln(2*2^64) ~= 45.0545
#define MAX_TANGENT_NORM 45.05455017f

typedef __attribute__((ext_vector_type(4))) float vf4;  // native vector for NT stores

__device__ __forceinline__ float wred(float v) {
#pragma unroll
    for (int m = 16; m >= 1; m >>= 1) v += __shfl_xor(v, m, 32);
    return v;
}

// K1: per (b, chunk of 128 rows) column sums, float4-wide (global_load_b128).
// grid = 64*8 blocks, 256 threads. Threads 0-127 handle rows [0,64), cols 4g..4g+3;
// threads 128-255 handle rows [64,128); halves combined through LDS.
__global__ __launch_bounds__(256) void lbn_partial(const float* __restrict__ x,
                                                   float* __restrict__ partial) {
    __shared__ float4 sh[128];
    const int b = blockIdx.x >> 3;
    const int c = blockIdx.x & 7;
    const int t = threadIdx.x;
    const int g = t & 127;  // column group: cols 4g..4g+3
    const int h = t >> 7;   // row half
    const float* base =
        x + ((size_t)b * 1024 + (size_t)c * 128 + (size_t)h * 64) * DLEN + 4 * g;
    float4 acc = make_float4(0.f, 0.f, 0.f, 0.f);
#pragma unroll 4
    for (int s = 0; s < 64; ++s) {
        float4 v = *(const float4*)(base + (size_t)s * DLEN);
        acc.x += v.x; acc.y += v.y; acc.z += v.z; acc.w += v.w;
    }
    if (h == 1) sh[g] = acc;
    __syncthreads();
    if (h == 0) {
        float4 o = sh[g];
        acc.x += o.x; acc.y += o.y; acc.z += o.z; acc.w += o.w;
        *(float4*)(partial + (size_t)blockIdx.x * DLEN + 4 * g) = acc;
    }
}

// K2: per-b centroid: combine 8 chunks, /1024, Lorentz-normalize. grid = 64.
__global__ __launch_bounds__(256) void lbn_cent(const float* __restrict__ partial,
                                                float* __restrict__ cent) {
    __shared__ float red[256];
    const int b = blockIdx.x;
    const int t = threadIdx.x;
    float a0 = 0.f, a1 = 0.f;
    for (int c = 0; c < 8; ++c) {
        const float* p = partial + (size_t)(b * 8 + c) * DLEN;
        a0 += p[t];
        a1 += p[t + 256];
    }
    a0 *= (1.0f / 1024.0f);
    a1 *= (1.0f / 1024.0f);
    // q = -linner(avg,avg) = avg0^2 - sum_{j>0} avg_j^2
    float q = -(a0 * a0) - (a1 * a1);
    if (t == 0) q += 2.0f * a0 * a0;
    red[t] = q;
    __syncthreads();
    for (int off = 128; off > 0; off >>= 1) {
        if (t < off) red[t] += red[t + off];
        __syncthreads();
    }
    const float denom = sqrtf(fmaxf(red[0], MIN_NORM));
    cent[(size_t)b * DLEN + t] = a0 / denom;
    cent[(size_t)b * DLEN + t + 256] = a1 / denom;
}

// K3: centroid over B=64 centroids -> mean[512]. grid = 1.
__global__ __launch_bounds__(256) void lbn_mean(const float* __restrict__ cent,
                                                float* __restrict__ meanv) {
    __shared__ float red[256];
    const int t = threadIdx.x;
    float a0 = 0.f, a1 = 0.f;
    for (int b = 0; b < 64; ++b) {
        const float* p = cent + (size_t)b * DLEN;
        a0 += p[t];
        a1 += p[t + 256];
    }
    a0 *= (1.0f / 64.0f);
    a1 *= (1.0f / 64.0f);
    float q = -(a0 * a0) - (a1 * a1);
    if (t == 0) q += 2.0f * a0 * a0;
    red[t] = q;
    __syncthreads();
    for (int off = 128; off > 0; off >>= 1) {
        if (t < off) red[t] += red[t + off];
        __syncthreads();
    }
    const float denom = sqrtf(fmaxf(red[0], MIN_NORM));
    meanv[t] = a0 / denom;
    meanv[t + 256] = a1 / denom;
}

// K4: main pass, one row (512 floats) per wave32, 8 waves/block.
__global__ __launch_bounds__(256) void lbn_main(
    const float* __restrict__ x, const float* __restrict__ meanv,
    const float* __restrict__ beta, const float* __restrict__ gamma,
    float* __restrict__ out, int nrows) {
    __shared__ float s_mean[DLEN];
    __shared__ float s_beta[DLEN];
    const int tid = threadIdx.x;
    const int lane = tid & 31;
    const int wv = tid >> 5;

    // CDNA5 async copy: stage mean (wave 0) and beta (wave 1) into LDS.
    // 4 x global_load_async_to_lds_b128 per wave = 2KB each (ASYNCcnt path).
    if (wv < 2) {
        const float* src = (wv == 0) ? meanv : beta;
        float* dst = (wv == 0) ? s_mean : s_beta;
#pragma unroll
        for (int c = 0; c < 4; ++c) {
            const int col = (c * 32 + lane) * 4;
            uint64_t ga = (uint64_t)(uintptr_t)(src + col);
            uint32_t lo = (uint32_t)(uintptr_t)(dst + col);  // LDS byte addr = low 32 bits
            asm volatile("global_load_async_to_lds_b128 %0, %1, off"
                         :: "v"(lo), "v"(ga) : "memory");
        }
        asm volatile("s_wait_asynccnt 0" ::: "memory");
    }
    __syncthreads();

    const int row = blockIdx.x * 8 + wv;
    if (row >= nrows) return;
    const float* xr = x + (size_t)row * DLEN;

    float xs[16], ms[16], bs[16];
#pragma unroll
    for (int c = 0; c < 4; ++c) {
        const int col = (c * 32 + lane) * 4;
        float4 tx = *(const float4*)(xr + col);
        float4 tm = *(const float4*)(s_mean + col);
        float4 tb = *(const float4*)(s_beta + col);
        xs[4 * c + 0] = tx.x; xs[4 * c + 1] = tx.y; xs[4 * c + 2] = tx.z; xs[4 * c + 3] = tx.w;
        ms[4 * c + 0] = tm.x; ms[4 * c + 1] = tm.y; ms[4 * c + 2] = tm.z; ms[4 * c + 3] = tm.w;
        bs[4 * c + 0] = tb.x; bs[4 * c + 1] = tb.y; bs[4 * c + 2] = tb.z; bs[4 * c + 3] = tb.w;
    }
    const bool l0 = (lane == 0);  // lane 0, i==0 holds the time coordinate j=0

    // Reduction 1: row sum and Lorentz inner <mean, x>_L
    float sx = 0.f, lmx = 0.f;
#pragma unroll
    for (int i = 0; i < 16; ++i) { sx += xs[i]; lmx += ms[i] * xs[i]; }
    if (l0) lmx -= 2.0f * ms[0] * xs[0];  // flip sign of time component
    sx = wred(sx);
    lmx = wred(lmx);

    const float alpha = fmaxf(-lmx, 1.0f + 1e-7f);
    const float dnm = sqrtf(alpha * alpha - 1.0f);
    const float coef = acoshf(alpha) / dnm;

    // Reduction 2: unbiased row variance of x (two-pass, matches jnp.var)
    const float mrow = sx * (1.0f / 512.0f);
    float ssd = 0.f;
#pragma unroll
    for (int i = 0; i < 16; ++i) { float d = xs[i] - mrow; ssd += d * d; }
    ssd = wred(ssd);
    const float stdv = sqrtf(ssd * (1.0f / 511.0f));
    const float s1 = gamma[0] / stdv + EPSV;

    // logmap + transp0back scalars
    const float m0 = s_mean[0];
    const float x0 = __shfl(xs[0], 0, 32);
    const float xT0 = coef * (x0 - alpha * m0);
    const float f1 = (-xT0) / (1.0f + m0);

    // v = (logmap transported to origin) * (gamma/std + eps); Reduction 3: ||v||^2
    float v[16];
    float n2 = 0.f;
#pragma unroll
    for (int i = 0; i < 16; ++i) {
        const float o = (l0 && i == 0) ? 1.0f : 0.0f;
        float t = coef * (xs[i] - alpha * ms[i]) + f1 * (ms[i] + o);
        t *= s1;
        v[i] = t;
        n2 += t * t;
    }
    n2 = wred(n2);
    const float nrm = sqrtf(fmaxf(n2, MIN_NORM));
    const float s2 = fminf(1.0f, MAX_TANGENT_NORM / nrm);

    // Reduction 4: <beta, v>_L for transp0
    float lbv = 0.f;
#pragma unroll
    for (int i = 0; i < 16; ++i) { v[i] *= s2; lbv += bs[i] * v[i]; }
    if (l0) lbv -= 2.0f * bs[0] * v[0];
    lbv = wred(lbv);
    const float b0 = s_beta[0];
    const float f2 = lbv / (1.0f + b0);

    // u = transp0(beta, v); Reduction 5: <u,u>_L for expmap
    float u[16];
    float luu = 0.f;
#pragma unroll
    for (int i = 0; i < 16; ++i) {
        const float o = (l0 && i == 0) ? 1.0f : 0.0f;
        const float t = v[i] + f2 * (bs[i] + o);
        u[i] = t;
        luu += t * t;
    }
    if (l0) luu -= 2.0f * u[0] * u[0];
    luu = wred(luu);
    const float nn = sqrtf(fmaxf(luu, MIN_NORM));
    const float ch = coshf(nn);
    const float sh = sinhf(nn) / nn;

    // Nontemporal stores: out is write-once; keep x resident in the 192MB L2.
    float* orow = out + (size_t)row * DLEN;
#pragma unroll
    for (int c = 0; c < 4; ++c) {
        const int col = (c * 32 + lane) * 4;
        vf4 r;
        r.x = ch * bs[4 * c + 0] + sh * u[4 * c + 0];
        r.y = ch * bs[4 * c + 1] + sh * u[4 * c + 1];
        r.z = ch * bs[4 * c + 2] + sh * u[4 * c + 2];
        r.w = ch * bs[4 * c + 3] + sh * u[4 * c + 3];
        __builtin_nontemporal_store(r, (vf4*)(orow + col));
    }
}

extern "C" void kernel_launch(void* const* d_in, const int* in_sizes, int n_in,
                              void* d_out, int out_size, void* d_ws, size_t ws_size,
                              hipStream_t stream) {
    const float* x = (const float*)d_in[0];      // [64,1024,512]
    const float* beta = (const float*)d_in[1];   // [512]
    const float* gamma = (const float*)d_in[2];  // [1]
    float* out = (float*)d_out;

    // Workspace layout: partial[512][512] | cent[64][512] | mean[512]
    float* partial = (float*)d_ws;
    float* cent = partial + 512 * DLEN;
    float* meanv = cent + 64 * DLEN;

    const int nrows = in_sizes[0] / DLEN;  // 65536

    lbn_partial<<<512, 256, 0, stream>>>(x, partial);
    lbn_cent<<<64, 256, 0, stream>>>(partial, cent);
    lbn_mean<<<1, 256, 0, stream>>>(cent, meanv);
    lbn_main<<<(nrows + 7) / 8, 256, 0, stream>>>(x, meanv, beta, gamma, out, nrows);
}